// RopeFinder_49529562857601
// MI455X (gfx1250) — compile-verified
//
#include <hip/hip_runtime.h>
#include <math.h>

#define GS     97            // plane is 97x97
#define NPIX   (GS*GS)       // 9409
#define PADN   112           // 7 * 16 tiles
#define VOL_D  150
#define VOL_H  512
#define VOL_W  512
#define FP_N   (NPIX*3)      // 28227 floats: f_p block / one n-block of grads

typedef __attribute__((ext_vector_type(2))) float v2f;
typedef __attribute__((ext_vector_type(8))) float v8f;

// 97x97 edge-corrected central-difference stencil matrix, zero-padded to 112.
// Branchless: pure compare+select chains -> v_cndmask, no exec divergence.
__device__ __forceinline__ float gmat(int r, int c) {
  // interior row: -0.5 at c==r-1, +0.5 at c==r+1
  float v = (c == r + 1) ? 0.5f : ((c == r - 1) ? -0.5f : 0.0f);
  // first row: [-1.5, 2.0, -0.5, 0, ...]
  float first = (c == 0) ? -1.5f : ((c == 1) ? 2.0f : ((c == 2) ? -0.5f : 0.0f));
  // last row: [..., 0.5, -2.0, 1.5]
  float last = (c == GS - 1) ? 1.5f
             : ((c == GS - 2) ? -2.0f : ((c == GS - 3) ? 0.5f : 0.0f));
  v = (r == 0) ? first : v;
  v = (r == GS - 1) ? last : v;
  v = (r >= GS || c >= GS) ? 0.0f : v;
  return v;
}

// Stage 1: per-pixel trilinear gather + vwn basis rotation (-> f_p in d_out),
// polar rotation (-> padded f_pol in ws), cos/sin(phi) cached in ws.
__global__ void sample_rotate_kernel(const float* __restrict__ vol,
                                     const float* __restrict__ pn,
                                     const float* __restrict__ org,
                                     float* __restrict__ out,
                                     float* __restrict__ fpol,
                                     float* __restrict__ cosP,
                                     float* __restrict__ sinP) {
  int t = blockIdx.x * blockDim.x + threadIdx.x;
  if (t >= NPIX) return;
  int h = t / GS, w = t % GS;

  // plane basis (uniform; matches plane_vw)
  float px = pn[0], py = pn[1], pz = pn[2];
  float nrm = sqrtf(px*px + py*py + pz*pz);
  float nx = px / nrm, ny = py / nrm, nz = pz / nrm;
  float beta = 1.0f / sqrtf(1.0f - nz*nz);
  float pvx = -ny * beta, pvy = nx * beta, pvz = 0.0f;
  float pwx = -nz * pvy,  pwy = nz * pvx,  pwz = 1.0f / beta;

  float a = (float)(w - 48);   // multiplies pv
  float b = (float)(h - 48);   // multiplies pw
  float gx_ = a*pvx + b*pwx + org[0];
  float gy_ = a*pvy + b*pwy + org[1];
  float gz_ = a*pvz + b*pwz + org[2];

  const float EPSF = 1e-8f;
  float gn0 = gx_ * (2.0f / ((float)(VOL_W - 1) + EPSF)) - 1.0f;
  float gn1 = gy_ * (2.0f / ((float)(VOL_H - 1) + EPSF)) - 1.0f;
  float gn2 = gz_ * (2.0f / ((float)(VOL_D - 1) + EPSF)) - 1.0f;

  float ix = ((gn0 + 1.0f) * (float)VOL_W - 1.0f) * 0.5f;
  float iy = ((gn1 + 1.0f) * (float)VOL_H - 1.0f) * 0.5f;
  float iz = ((gn2 + 1.0f) * (float)VOL_D - 1.0f) * 0.5f;

  float x0f = floorf(ix), y0f = floorf(iy), z0f = floorf(iz);
  float wx = ix - x0f, wy = iy - y0f, wz = iz - z0f;
  int x0 = (int)x0f, y0 = (int)y0f, z0 = (int)z0f;

  const size_t CH = (size_t)VOL_D * VOL_H * VOL_W;
  float s0 = 0.0f, s1 = 0.0f, s2 = 0.0f;
  #pragma unroll
  for (int dz = 0; dz < 2; ++dz) {
    int zc = z0 + dz;
    float wz_ = dz ? wz : (1.0f - wz);
    #pragma unroll
    for (int dy = 0; dy < 2; ++dy) {
      int yc = y0 + dy;
      float wy_ = dy ? wy : (1.0f - wy);
      #pragma unroll
      for (int dx = 0; dx < 2; ++dx) {
        int xc = x0 + dx;
        float wx_ = dx ? wx : (1.0f - wx);
        bool valid = (xc >= 0) & (xc < VOL_W) & (yc >= 0) & (yc < VOL_H) &
                     (zc >= 0) & (zc < VOL_D);
        int xs = xc < 0 ? 0 : (xc > VOL_W-1 ? VOL_W-1 : xc);
        int ys = yc < 0 ? 0 : (yc > VOL_H-1 ? VOL_H-1 : yc);
        int zs = zc < 0 ? 0 : (zc > VOL_D-1 ? VOL_D-1 : zc);
        float wgt = wz_ * wy_ * wx_ * (valid ? 1.0f : 0.0f);
        size_t base = (size_t)zs * (VOL_H * VOL_W) + (size_t)ys * VOL_W + (size_t)xs;
        s0 += vol[base]          * wgt;
        s1 += vol[base +   CH]   * wgt;
        s2 += vol[base + 2*CH]   * wgt;
      }
    }
  }

  // vwn rotation: rows [pv; pw; n]
  float fp0 = pvx*s0 + pvy*s1 + pvz*s2;
  float fp1 = pwx*s0 + pwy*s1 + pwz*s2;
  float fp2 = nx*s0  + ny*s1  + nz*s2;

  out[0*NPIX + t] = fp0;   // f_p output (1,3,1,97,97)
  out[1*NPIX + t] = fp1;
  out[2*NPIX + t] = fp2;

  // polar rotation P(phi)
  float phi = atan2f((float)(h - 48), (float)(w - 48));
  float cc = cosf(phi), ss = sinf(phi);
  cosP[t] = cc;
  sinP[t] = ss;
  size_t pp = (size_t)h * PADN + w;
  fpol[0*PADN*PADN + pp] =  cc*fp0 + ss*fp1;
  fpol[1*PADN*PADN + pp] = -ss*fp0 + cc*fp1;
  fpol[2*PADN*PADN + pp] =  fp2;
}

// Stage 2: gradients as fp32 WMMA GEMMs with band-limited K.
//   gx = F * G^T  (grad along W),  gy = G * F  (grad along H), then rotate by P.
// G is tridiagonal (+3-entry edge rows), so for a 16-row block of G only
// K in [lo-2, lo+16] contributes; all other k-blocks are exactly zero.
// One wave (32 lanes) per (channel, 16x16 output tile): grid = 3*7*7 = 147.
__global__ __launch_bounds__(32)
void grad_wmma_kernel(const float* __restrict__ fpol,
                      const float* __restrict__ cosP,
                      const float* __restrict__ sinP,
                      float* __restrict__ out) {
  int bid  = blockIdx.x;      // 0..146
  int ch   = bid / 49;
  int tile = bid % 49;
  int tm   = tile / 7;        // M-tile (rows h)
  int tn   = tile % 7;        // N-tile (cols w)
  int lane = threadIdx.x;     // wave32
  int half = lane >> 4;       // 0: K=k0,k0+1   1: K=k0+2,k0+3
  int l15  = lane & 15;
  int koff = half * 2;

  const float* F = fpol + (size_t)ch * (PADN * PADN);
  int rowA = tm * 16 + l15;   // h row (gx A rows / gy G rows)
  int colB = tn * 16 + l15;   // w col (gx G^T cols / gy F cols)

  v8f accX = {0.f,0.f,0.f,0.f,0.f,0.f,0.f,0.f};
  v8f accY = {0.f,0.f,0.f,0.f,0.f,0.f,0.f,0.f};

  // gx: B side is G rows [tn*16, tn*16+15] -> K band around tn.
  {
    int lo   = tn * 16;
    int kmin = lo - 2 < 0 ? 0 : lo - 2;
    int kmax = lo + 16 > 96 ? 96 : lo + 16;
    for (int k0 = kmin & ~3; k0 <= kmax; k0 += 4) {
      int k = k0 + koff;
      v2f aX = *(const v2f*)(F + (size_t)rowA * PADN + k);   // contiguous b64
      v2f bX; bX.x = gmat(colB, k);  bX.y = gmat(colB, k + 1);
      accX = __builtin_amdgcn_wmma_f32_16x16x4_f32(false, aX, false, bX,
                                                   (short)0, accX, false, false);
    }
  }
  // gy: A side is G rows [tm*16, tm*16+15] -> K band around tm.
  {
    int lo   = tm * 16;
    int kmin = lo - 2 < 0 ? 0 : lo - 2;
    int kmax = lo + 16 > 96 ? 96 : lo + 16;
    for (int k0 = kmin & ~3; k0 <= kmax; k0 += 4) {
      int k = k0 + koff;
      v2f aY; aY.x = gmat(rowA, k);  aY.y = gmat(rowA, k + 1);
      v2f bY; bY.x = F[(size_t)k * PADN + colB];             // stride-PADN pair
              bY.y = F[(size_t)(k + 1) * PADN + colB];
      accY = __builtin_amdgcn_wmma_f32_16x16x4_f32(false, aY, false, bY,
                                                   (short)0, accY, false, false);
    }
  }

  // D layout: VGPR v -> M = v (+8 for lanes 16-31), N = lane (mod 16).
  #pragma unroll
  for (int v = 0; v < 8; ++v) {
    int hh = tm * 16 + v + half * 8;
    int ww = tn * 16 + l15;
    if (hh < GS && ww < GS) {
      int p = hh * GS + ww;
      float cc = cosP[p], ss = sinP[p];
      float gx = accX[v], gy = accY[v];
      // grads_axial (1,3,3,1,97,97): 28227 + n*28227 + i*9409 + h*97 + w, i = ch
      size_t base = (size_t)FP_N + (size_t)ch * NPIX + (size_t)hh * GS + ww;
      out[base]            =  cc * gx + ss * gy;  // n = 0
      out[base +   FP_N]   = -ss * gx + cc * gy;  // n = 1
      out[base + 2*FP_N]   =  0.0f;               // n = 2 (P[2]=(0,0,1), gz=0)
    }
  }
}

extern "C" void kernel_launch(void* const* d_in, const int* in_sizes, int n_in,
                              void* d_out, int out_size, void* d_ws, size_t ws_size,
                              hipStream_t stream) {
  const float* vol = (const float*)d_in[0];   // (1,3,150,512,512) f32
  const float* pn  = (const float*)d_in[1];   // (3,) f32
  const float* org = (const float*)d_in[2];   // (3,) f32
  float* out = (float*)d_out;                 // 28227 + 84681 floats
  float* ws  = (float*)d_ws;

  float* fpol = ws;                           // 3 * 112 * 112 (zero-padded)
  float* cosP = ws + 3 * PADN * PADN;         // 9409
  float* sinP = cosP + NPIX;                  // 9409

  hipMemsetAsync(fpol, 0, (size_t)3 * PADN * PADN * sizeof(float), stream);
  sample_rotate_kernel<<<(NPIX + 255) / 256, 256, 0, stream>>>(
      vol, pn, org, out, fpol, cosP, sinP);
  grad_wmma_kernel<<<147, 32, 0, stream>>>(fpol, cosP, sinP, out);
}